// PostProcess_54082228191709
// MI455X (gfx1250) — compile-verified
//
#include <hip/hip_runtime.h>

// Detection post-process for MI455X (gfx1250):
//   K1: TDM-staged class-logit reduction (max/argmax/sigmoid) + box decode/clip
//   K2: greedy NMS, one workgroup per image
//
// CDNA5-specific paths used: tensor_load_to_lds (TDM) + s_wait_tensorcnt.

namespace {
constexpr int   kC          = 90;
constexpr int   kMaxDet     = 100;
constexpr float kConfThr    = 0.25f;
constexpr float kIouThr     = 0.45f;
constexpr float kImg        = 512.0f;
constexpr int   kAPB        = 128;    // anchors per block in score kernel (4 waves x 32)
constexpr int   kNmsThreads = 1024;
}

typedef unsigned int u32x4 __attribute__((ext_vector_type(4)));
typedef int          i32x8 __attribute__((ext_vector_type(8)));
typedef int          i32x4 __attribute__((ext_vector_type(4)));

// ---------------------------------------------------------------------------
// Kernel 1: per-anchor class reduction (via TDM -> LDS) + bbox transform/clip
// ---------------------------------------------------------------------------
__global__ __launch_bounds__(kAPB)
void score_box_kernel(const float* __restrict__ reg,   // [B,N,4]
                      const float* __restrict__ cls,   // [B,N,C]
                      const float* __restrict__ anc,   // [1,N,4] (y1,x1,y2,x2)
                      float* __restrict__ boxes,       // [B,N,4] out (x1,y1,x2,y2)
                      float* __restrict__ maxscore,    // [B,N] out
                      int*   __restrict__ label,       // [B,N] out
                      float* __restrict__ sc,          // [B,N] out (thresholded)
                      int N) {
  __shared__ float tile[kAPB * kC];                    // 128 * 90 * 4B = 45 KB

  const int b    = (int)blockIdx.y;
  const int n0   = (int)blockIdx.x * kAPB;
  const int wave = (int)__builtin_amdgcn_readfirstlane(threadIdx.x >> 5);
  const int lane = (int)(threadIdx.x & 31);
  const int nw   = n0 + wave * 32;                     // first anchor of this wave
  const int rows = N - nw;                             // anchors remaining from nw

  if (rows > 0) {
    // ---- Tensor DMA descriptor (D#), 2D tile: 90 x 32 elements of 4 bytes ----
    const unsigned tdim1 = (unsigned)rows;             // OOB rows read as zero
    const unsigned lds   = (unsigned)(unsigned long long)(void*)&tile[wave * 32 * kC];
    const unsigned long long ga =
        (unsigned long long)(const void*)(cls + ((unsigned long long)b * (unsigned)N
                                                 + (unsigned)nw) * kC);
    // Group 0: count=1 | lds_addr | global_addr[56:0] | type=2
    u32x4 g0 = { 1u,
                 lds,
                 (unsigned)(ga & 0xFFFFFFFFull),
                 (unsigned)((ga >> 32) & 0x1FFFFFFull) | (2u << 30) };
    // Group 1: data_size=2 (4B); tensor_dim0=90; tensor_dim1=rows;
    //          tile_dim0=90; tile_dim1=32; tile_dim2=0; dim0_stride=90
    i32x8 g1 = { (int)(2u << 16),
                 (int)((unsigned)kC << 16),
                 (int)((tdim1 & 0xFFFFu) << 16),
                 (int)((tdim1 >> 16) | ((unsigned)kC << 16)),
                 32,
                 kC,
                 0, 0 };
    i32x4 zero4 = { 0, 0, 0, 0 };
    i32x8 zero8 = { 0, 0, 0, 0, 0, 0, 0, 0 };
    __builtin_amdgcn_tensor_load_to_lds(g0, g1, zero4, zero4, zero8, 0);
  }
  __builtin_amdgcn_s_wait_tensorcnt(0);
  asm volatile("" ::: "memory");

  const int n = nw + lane;
  if (n >= N) return;

  // ---- max/argmax over classes from LDS (first-max index like jnp.argmax) ----
  const float* row = &tile[(wave * 32 + lane) * kC];
  float m = row[0];
  int lab = 0;
#pragma unroll 10
  for (int c = 1; c < kC; ++c) {
    float v = row[c];
    if (v > m) { m = v; lab = c; }
  }
  const float sig = 1.0f / (1.0f + __expf(-m));

  const unsigned long long idx = (unsigned long long)b * (unsigned)N + (unsigned)n;
  maxscore[idx] = sig;
  label[idx]    = lab;
  sc[idx]       = (sig > kConfThr) ? sig : -1.0f;

  // ---- bbox transform + clip (anchors are (y1,x1,y2,x2); out (x1,y1,x2,y2)) ----
  const float4 a = ((const float4*)anc)[n];
  const float4 r = ((const float4*)reg)[idx];          // (dy,dx,dh,dw)
  const float ycA = 0.5f * (a.x + a.z);
  const float xcA = 0.5f * (a.y + a.w);
  const float ha  = a.z - a.x;
  const float wa  = a.w - a.y;
  const float w   = __expf(r.w) * wa;
  const float h   = __expf(r.z) * ha;
  const float yc  = r.x * ha + ycA;
  const float xc  = r.y * wa + xcA;
  float x1 = xc - 0.5f * w, y1 = yc - 0.5f * h;
  float x2 = xc + 0.5f * w, y2 = yc + 0.5f * h;
  x1 = fminf(fmaxf(x1, 0.0f), kImg);
  x2 = fminf(fmaxf(x2, 0.0f), kImg);
  y1 = fminf(fmaxf(y1, 0.0f), kImg);
  y2 = fminf(fmaxf(y2, 0.0f), kImg);
  ((float4*)boxes)[idx] = make_float4(x1, y1, x2, y2);
}

// ---------------------------------------------------------------------------
// Kernel 2: greedy NMS, one 1024-thread workgroup per image
// ---------------------------------------------------------------------------
__global__ __launch_bounds__(kNmsThreads)
void nms_kernel(const float* __restrict__ boxes,      // [B,N,4]
                const float* __restrict__ maxscore,   // [B,N]
                const int*   __restrict__ label,      // [B,N]
                float* __restrict__ sc,               // [B,N] (mutated)
                float* __restrict__ out,              // scores | labels | boxes
                int B, int N) {
  const int b = (int)blockIdx.x;
  const int t = (int)threadIdx.x;

  const float* bx = boxes + (unsigned long long)b * (unsigned)N * 4u;
  float* s = sc + (unsigned long long)b * (unsigned)N;
  float* out_scores = out + (unsigned long long)b * kMaxDet;
  float* out_labels = out + (unsigned long long)B * kMaxDet
                          + (unsigned long long)b * kMaxDet;
  float* out_boxes  = out + 2ull * (unsigned long long)B * kMaxDet
                          + (unsigned long long)b * kMaxDet * 4u;

  __shared__ float rv[kNmsThreads];
  __shared__ int   ri[kNmsThreads];
  __shared__ float bbx[4];
  __shared__ int   bidx;
  __shared__ float bval;

  for (int k = 0; k < kMaxDet; ++k) {
    // ---- block-wide argmax over sc (ties -> smallest index, like jnp.argmax)
    float best = -3.4e38f;
    int   bi   = 0x7FFFFFFF;
    for (int n = t; n < N; n += kNmsThreads) {
      float v = s[n];
      if (v > best) { best = v; bi = n; }
    }
    rv[t] = best;
    ri[t] = bi;
    __syncthreads();
    for (int off = kNmsThreads / 2; off > 0; off >>= 1) {
      if (t < off) {
        float v2 = rv[t + off];
        int   i2 = ri[t + off];
        if (v2 > rv[t] || (v2 == rv[t] && i2 < ri[t])) { rv[t] = v2; ri[t] = i2; }
      }
      __syncthreads();
    }
    if (t == 0) {
      bidx = ri[0];
      bval = rv[0];
      if (rv[0] > 0.0f) {
        const float* p = bx + (unsigned long long)ri[0] * 4u;
        bbx[0] = p[0]; bbx[1] = p[1]; bbx[2] = p[2]; bbx[3] = p[3];
      }
    }
    __syncthreads();

    if (!(bval > 0.0f)) {
      // no more valid detections: fill the remaining padded slots
      for (int kk = k + t; kk < kMaxDet; kk += kNmsThreads) {
        out_scores[kk] = 0.0f;
        out_labels[kk] = -1.0f;
        out_boxes[(unsigned long long)kk * 4u + 0] = 0.0f;
        out_boxes[(unsigned long long)kk * 4u + 1] = 0.0f;
        out_boxes[(unsigned long long)kk * 4u + 2] = 0.0f;
        out_boxes[(unsigned long long)kk * 4u + 3] = 0.0f;
      }
      break;
    }

    const int bidx_l = bidx;
    if (t == 0) {
      const unsigned long long gi = (unsigned long long)b * (unsigned)N
                                  + (unsigned)bidx_l;
      out_scores[k] = maxscore[gi];
      out_labels[k] = (float)label[gi];
      out_boxes[(unsigned long long)k * 4u + 0] = bbx[0];
      out_boxes[(unsigned long long)k * 4u + 1] = bbx[1];
      out_boxes[(unsigned long long)k * 4u + 2] = bbx[2];
      out_boxes[(unsigned long long)k * 4u + 3] = bbx[3];
    }

    // ---- suppression: IoU(best, all) > thr  -> sc = -1 (self included) ----
    const float x1b = bbx[0], y1b = bbx[1], x2b = bbx[2], y2b = bbx[3];
    const float area_b = (x2b - x1b) * (y2b - y1b);
    for (int n = t; n < N; n += kNmsThreads) {
      const float4 q = ((const float4*)bx)[n];
      const float xx1 = fmaxf(x1b, q.x);
      const float yy1 = fmaxf(y1b, q.y);
      const float xx2 = fminf(x2b, q.z);
      const float yy2 = fminf(y2b, q.w);
      const float inter = fmaxf(xx2 - xx1, 0.0f) * fmaxf(yy2 - yy1, 0.0f);
      const float areas = (q.z - q.x) * (q.w - q.y);
      const float iou = inter / (area_b + areas - inter + 1e-8f);
      if (iou > kIouThr) s[n] = -1.0f;
    }
    if (t == 0) s[bidx_l] = -1.0f;
    __syncthreads();   // HIP __syncthreads includes workgroup-scope mem fences
  }
}

// ---------------------------------------------------------------------------
// Host launch
// ---------------------------------------------------------------------------
extern "C" void kernel_launch(void* const* d_in, const int* in_sizes, int n_in,
                              void* d_out, int out_size, void* d_ws, size_t ws_size,
                              hipStream_t stream) {
  const float* reg = (const float*)d_in[0];   // [B,N,4]
  const float* cls = (const float*)d_in[1];   // [B,N,C]
  const float* anc = (const float*)d_in[2];   // [1,N,4]
  // d_in[3] = imgs; only shape (512,512) is used (hardcoded as kImg)

  const int N = in_sizes[2] / 4;              // anchors: 1*N*4
  const int B = in_sizes[0] / (N * 4);        // regression: B*N*4

  // workspace layout (floats): boxes[B*N*4] | maxscore[B*N] | label[B*N] | sc[B*N]
  float* boxes    = (float*)d_ws;
  float* maxscore = boxes + (size_t)B * N * 4;
  int*   label    = (int*)(maxscore + (size_t)B * N);
  float* sc       = (float*)((int*)label + (size_t)B * N);

  dim3 grid1((N + kAPB - 1) / kAPB, B);
  score_box_kernel<<<grid1, kAPB, 0, stream>>>(reg, cls, anc,
                                               boxes, maxscore, label, sc, N);

  nms_kernel<<<B, kNmsThreads, 0, stream>>>(boxes, maxscore, label, sc,
                                            (float*)d_out, B, N);
}